// RWKV_RNN_55980603736370
// MI455X (gfx1250) — compile-verified
//
#include <hip/hip_runtime.h>
#include <math.h>

#define L_ 24
#define D_ 1024
#define H_ 4096
#define V_ 50277

typedef __attribute__((ext_vector_type(2))) float v2f;
typedef __attribute__((ext_vector_type(4))) float v4f;
typedef __attribute__((ext_vector_type(8))) float v8f;

#define MV_ADD    1
#define MV_GATE   2
#define MV_RELUSQ 4

// ---------------- block-wide sum / sumsq reduction (256 threads) -------------
__device__ __forceinline__ void reduce2(float s, float s2, float* S, float* S2) {
    __shared__ float r1[256], r2[256];
    int tid = threadIdx.x;
    r1[tid] = s; r2[tid] = s2;
    __syncthreads();
    for (int off = 128; off > 0; off >>= 1) {
        if (tid < off) { r1[tid] += r1[tid + off]; r2[tid] += r2[tid + off]; }
        __syncthreads();
    }
    *S = r1[0]; *S2 = r2[0];
}

// ---------------- WMMA f32 streaming matvec core (one wave = 16 rows) --------
// y[r] = (mode&ADD ? addv[r] : 0) + gate(r) * act(sum_k W[r,k]*xs[k])
// A(16x4 f32): lanes0-15 = logical K{0,1}, lanes16-31 = logical K{2,3},
// M = lane&15. Lane (h=lane>>4, r=lane&15) nontemporal-loads b128
// W[r, k+4h..k+4h+3]; two wmmas per 8 physical K. x broadcast across all N
// columns of B so every column of D equals y.
__device__ __forceinline__ void mv_core(const float* __restrict__ W,
                                        const float* __restrict__ xs,
                                        float* __restrict__ y,
                                        int nrows, int ncols, int row0,
                                        int mode,
                                        const float* __restrict__ gate,
                                        const float* __restrict__ addv) {
    int lane = threadIdx.x & 31;
    int half = lane >> 4;                       // 0 or 1
    int r    = lane & 15;
    int row  = row0 + r;
    int rl   = row < nrows ? row : nrows - 1;   // clamp tail loads, mask writes
    const v4f* wrow = (const v4f*)(W + (size_t)rl * ncols);

    v8f c = {};
    #pragma unroll 4
    for (int k = 0; k < ncols; k += 8) {
        v4f f = __builtin_nontemporal_load(wrow + (k >> 2) + half);
        float2 bA = *(const float2*)(xs + k + 4 * half);
        float2 bB = *(const float2*)(xs + k + 4 * half + 2);
        v2f a1 = { f.x, f.y };
        v2f b1 = { bA.x, bA.y };
        c = __builtin_amdgcn_wmma_f32_16x16x4_f32(false, a1, false, b1,
                                                  (short)0, c, false, false);
        v2f a2 = { f.z, f.w };
        v2f b2 = { bB.x, bB.y };
        c = __builtin_amdgcn_wmma_f32_16x16x4_f32(false, a2, false, b2,
                                                  (short)0, c, false, false);
    }

    // C/D layout: vgpr g, lanes0-15 -> M=g ; lanes16-31 -> M=8+g (all N equal)
    if (r == 0) {
        for (int g = 0; g < 8; ++g) {
            int orow = row0 + 8 * half + g;
            if (orow < nrows) {
                float d = c[g];
                if (mode & MV_RELUSQ) d = d > 0.f ? d * d : 0.f;
                float gv = (mode & MV_GATE)
                               ? 1.f / (1.f + __expf(-gate[orow])) : 1.f;
                float base = (mode & MV_ADD) ? addv[orow] : 0.f;
                y[orow] = base + gv * d;
            }
        }
    }
}

// ---------------- embed + ln0 (one block) ------------------------------------
__global__ void __launch_bounds__(256)
k_embed_ln(const int* __restrict__ tok, const float* __restrict__ emb,
           const float* __restrict__ w, const float* __restrict__ b,
           float* __restrict__ xout) {
    int t = tok[0];
    const float* src = emb + (size_t)t * D_;
    float loc[4], s = 0.f, s2 = 0.f;
    for (int j = 0; j < 4; ++j) {
        int i = threadIdx.x + j * 256;
        float v = src[i]; loc[j] = v; s += v; s2 += v * v;
    }
    float S, S2; reduce2(s, s2, &S, &S2);
    float mu = S / D_, var = S2 / D_ - mu * mu;
    float rs = rsqrtf(var + 1e-5f);
    for (int j = 0; j < 4; ++j) {
        int i = threadIdx.x + j * 256;
        xout[i] = (loc[j] - mu) * rs * w[i] + b[i];
    }
}

// ---------------- fused LN1 + time-mix + k/v/r matvecs (24 blocks) -----------
__global__ void __launch_bounds__(256)
k_mv3(const float* __restrict__ x, const float* __restrict__ w,
      const float* __restrict__ b, const float* __restrict__ mk,
      const float* __restrict__ mvv, const float* __restrict__ mr,
      const float* __restrict__ sx,          // state row 1 (sx_att)
      float* __restrict__ st1_out,           // new state row 1 = xn
      const float* __restrict__ kw, const float* __restrict__ vw,
      const float* __restrict__ rw,
      float* __restrict__ kout, float* __restrict__ vout,
      float* __restrict__ rout) {
    __shared__ float xs[D_];
    float loc[4], s = 0.f, s2 = 0.f;
    for (int j = 0; j < 4; ++j) {
        int i = threadIdx.x + j * 256;
        float v = x[i]; loc[j] = v; s += v; s2 += v * v;
    }
    float S, S2; reduce2(s, s2, &S, &S2);
    float mu = S / D_, var = S2 / D_ - mu * mu;
    float rstd = rsqrtf(var + 1e-5f);

    int job = blockIdx.x >> 3;                           // 0:k 1:v 2:r
    const float* m = (job == 0) ? mk : (job == 1) ? mvv : mr;
    for (int j = 0; j < 4; ++j) {
        int i = threadIdx.x + j * 256;
        float xn = (loc[j] - mu) * rstd * w[i] + b[i];
        if (blockIdx.x == 0) st1_out[i] = xn;
        float mm = m[i];
        xs[i] = xn * mm + sx[i] * (1.f - mm);
    }
    __syncthreads();

    const float* W = (job == 0) ? kw : (job == 1) ? vw : rw;
    float*       y = (job == 0) ? kout : (job == 1) ? vout : rout;
    int wave = threadIdx.x >> 5;
    int row0 = ((blockIdx.x & 7) * 8 + wave) * 16;
    mv_core(W, xs, y, D_, D_, row0, 0, nullptr, nullptr);
}

// ---------------- fused WKV + output matvec + residual (8 blocks) ------------
__global__ void __launch_bounds__(256)
k_wkv_ow(const float* __restrict__ kv, const float* __restrict__ vv,
         const float* __restrict__ rpre, const float* __restrict__ tf,
         const float* __restrict__ td, const float* __restrict__ aa_in,
         const float* __restrict__ bb_in, const float* __restrict__ pp_in,
         float* __restrict__ aa_out, float* __restrict__ bb_out,
         float* __restrict__ qq_out,
         const float* __restrict__ ow, float* __restrict__ x /* in/out */) {
    __shared__ float xs[D_];
    for (int j = 0; j < 4; ++j) {
        int i = threadIdx.x + j * 256;
        float k = kv[i], v = vv[i];
        float aa = aa_in[i], bb = bb_in[i], pp = pp_in[i];
        float r = 1.f / (1.f + __expf(-rpre[i]));
        float ww = tf[i] + k;
        float qq = fmaxf(pp, ww);
        float e1 = __expf(pp - qq), e2 = __expf(ww - qq);
        float wkv = (e1 * aa + e2 * v) / (e1 * bb + e2);
        xs[i] = r * wkv;
        if (blockIdx.x == 0) {
            float ww2 = pp + td[i];
            float qq2 = fmaxf(ww2, k);
            float e1b = __expf(ww2 - qq2), e2b = __expf(k - qq2);
            aa_out[i] = e1b * aa + e2b * v;
            bb_out[i] = e1b * bb + e2b;
            qq_out[i] = qq2;
        }
    }
    __syncthreads();
    int wave = threadIdx.x >> 5;
    int row0 = (blockIdx.x * 8 + wave) * 16;
    mv_core(ow, xs, x, D_, D_, row0, MV_ADD, nullptr, x);
}

// ---------------- fused LN2 + mix + fkw(relu^2)/frw matvecs (40 blocks) ------
__global__ void __launch_bounds__(256)
k_ffn_kr(const float* __restrict__ x, const float* __restrict__ w,
         const float* __restrict__ b, const float* __restrict__ fmk,
         const float* __restrict__ fmr,
         const float* __restrict__ sx,         // state row 0 (sx_ffn)
         float* __restrict__ st0_out,          // new state row 0 = xn2
         const float* __restrict__ fkw, const float* __restrict__ frw,
         float* __restrict__ kkout, float* __restrict__ r2out) {
    __shared__ float xs[D_];
    float loc[4], s = 0.f, s2 = 0.f;
    for (int j = 0; j < 4; ++j) {
        int i = threadIdx.x + j * 256;
        float v = x[i]; loc[j] = v; s += v; s2 += v * v;
    }
    float S, S2; reduce2(s, s2, &S, &S2);
    float mu = S / D_, var = S2 / D_ - mu * mu;
    float rstd = rsqrtf(var + 1e-5f);

    int job = (blockIdx.x < 32) ? 0 : 1;                 // 0:fkw 1:frw
    const float* m = job ? fmr : fmk;
    for (int j = 0; j < 4; ++j) {
        int i = threadIdx.x + j * 256;
        float xn = (loc[j] - mu) * rstd * w[i] + b[i];
        if (blockIdx.x == 0) st0_out[i] = xn;
        float mm = m[i];
        xs[i] = xn * mm + sx[i] * (1.f - mm);
    }
    __syncthreads();

    int wave = threadIdx.x >> 5;
    if (job == 0) {
        int row0 = (blockIdx.x * 8 + wave) * 16;         // 4096 rows
        mv_core(fkw, xs, kkout, H_, D_, row0, MV_RELUSQ, nullptr, nullptr);
    } else {
        int row0 = ((blockIdx.x - 32) * 8 + wave) * 16;  // 1024 rows
        mv_core(frw, xs, r2out, D_, D_, row0, 0, nullptr, nullptr);
    }
}

// ---------------- fvw matvec, sigmoid gate + residual (8 blocks) -------------
__global__ void __launch_bounds__(256)
k_fvw(const float* __restrict__ kk, const float* __restrict__ r2pre,
      const float* __restrict__ fvw, float* __restrict__ x /* in/out */) {
    __shared__ float xs[H_];
    for (int i = threadIdx.x; i < H_; i += 256) xs[i] = kk[i];
    __syncthreads();
    int wave = threadIdx.x >> 5;
    int row0 = (blockIdx.x * 8 + wave) * 16;
    mv_core(fvw, xs, x, D_, H_, row0, MV_ADD | MV_GATE, r2pre, x);
}

// ---------------- fused LN-out + head matvec (393 blocks) --------------------
__global__ void __launch_bounds__(256)
k_head(const float* __restrict__ x, const float* __restrict__ w,
       const float* __restrict__ b, const float* __restrict__ head,
       float* __restrict__ logits) {
    __shared__ float xs[D_];
    float loc[4], s = 0.f, s2 = 0.f;
    for (int j = 0; j < 4; ++j) {
        int i = threadIdx.x + j * 256;
        float v = x[i]; loc[j] = v; s += v; s2 += v * v;
    }
    float S, S2; reduce2(s, s2, &S, &S2);
    float mu = S / D_, var = S2 / D_ - mu * mu;
    float rstd = rsqrtf(var + 1e-5f);
    for (int j = 0; j < 4; ++j) {
        int i = threadIdx.x + j * 256;
        xs[i] = (loc[j] - mu) * rstd * w[i] + b[i];
    }
    __syncthreads();
    int wave = threadIdx.x >> 5;
    int row0 = (blockIdx.x * 8 + wave) * 16;
    if (row0 < V_)
        mv_core(head, xs, logits, V_, D_, row0, 0, nullptr, nullptr);
}

// ---------------- host-side orchestration ------------------------------------
extern "C" void kernel_launch(void* const* d_in, const int* in_sizes, int n_in,
                              void* d_out, int out_size, void* d_ws, size_t ws_size,
                              hipStream_t stream) {
    (void)in_sizes; (void)n_in; (void)out_size; (void)ws_size;
    const int*   tok        = (const int*)d_in[0];
    const float* state      = (const float*)d_in[1];
    const float* emb        = (const float*)d_in[2];
    const float* ln0_w      = (const float*)d_in[3];
    const float* ln0_b      = (const float*)d_in[4];
    const float* ln1_w      = (const float*)d_in[5];
    const float* ln1_b      = (const float*)d_in[6];
    const float* ln2_w      = (const float*)d_in[7];
    const float* ln2_b      = (const float*)d_in[8];
    const float* att_key    = (const float*)d_in[9];
    const float* att_value  = (const float*)d_in[10];
    const float* att_recept = (const float*)d_in[11];
    const float* att_output = (const float*)d_in[12];
    const float* tm_k       = (const float*)d_in[13];
    const float* tm_v       = (const float*)d_in[14];
    const float* tm_r       = (const float*)d_in[15];
    const float* time_first = (const float*)d_in[16];
    const float* time_decay = (const float*)d_in[17];
    const float* ffn_key    = (const float*)d_in[18];
    const float* ffn_value  = (const float*)d_in[19];
    const float* ffn_recept = (const float*)d_in[20];
    const float* ffn_tm_k   = (const float*)d_in[21];
    const float* ffn_tm_r   = (const float*)d_in[22];
    const float* lnout_w    = (const float*)d_in[23];
    const float* lnout_b    = (const float*)d_in[24];
    const float* head       = (const float*)d_in[25];

    float* logits = (float*)d_out;             // [V]
    float* stout  = (float*)d_out + V_;        // [L,5,D]

    float* ws    = (float*)d_ws;               // 9*D floats = 36 KB scratch
    float* x     = ws + 0 * D_;
    float* kbuf  = ws + 1 * D_;
    float* vbuf  = ws + 2 * D_;
    float* rpre  = ws + 3 * D_;
    float* r2pre = ws + 4 * D_;
    float* kk    = ws + 5 * D_;                // H = 4*D floats

    k_embed_ln<<<1, 256, 0, stream>>>(tok, emb, ln0_w, ln0_b, x);

    for (int l = 0; l < L_; ++l) {
        const float* sl = state + (size_t)l * 5 * D_;
        float*       ol = stout + (size_t)l * 5 * D_;
        size_t dd = (size_t)l * D_ * D_;
        size_t hd = (size_t)l * H_ * D_;

        k_mv3<<<24, 256, 0, stream>>>(x, ln1_w + l * D_, ln1_b + l * D_,
                                      tm_k + l * D_, tm_v + l * D_,
                                      tm_r + l * D_, sl + 1 * D_, ol + 1 * D_,
                                      att_key + dd, att_value + dd,
                                      att_recept + dd, kbuf, vbuf, rpre);

        k_wkv_ow<<<8, 256, 0, stream>>>(kbuf, vbuf, rpre,
                                        time_first + l * D_, time_decay + l * D_,
                                        sl + 2 * D_, sl + 3 * D_, sl + 4 * D_,
                                        ol + 2 * D_, ol + 3 * D_, ol + 4 * D_,
                                        att_output + dd, x);

        k_ffn_kr<<<40, 256, 0, stream>>>(x, ln2_w + l * D_, ln2_b + l * D_,
                                         ffn_tm_k + l * D_, ffn_tm_r + l * D_,
                                         sl + 0 * D_, ol + 0 * D_,
                                         ffn_key + hd, ffn_recept + dd,
                                         kk, r2pre);

        k_fvw<<<8, 256, 0, stream>>>(kk, r2pre, ffn_value + hd, x);
    }

    k_head<<<393, 256, 0, stream>>>(x, lnout_w, lnout_b, head, logits);
}